// VolSDFVolumeRendering_30468497998289
// MI455X (gfx1250) — compile-verified
//
#include <hip/hip_runtime.h>

// VolSDF volume rendering weights for MI455X (gfx1250, wave32).
// Memory-bound (~201 MB @ 23.3 TB/s ~= 8.6 us floor). The per-ray exclusive
// cumsum is done as lower-triangular matmuls on V_WMMA_F32_16X16X4_F32
// (full fp32 accumulation), 16 rays per wave.

#define N_SAMPLES       128
#define RAYS_PER_WAVE   16
#define WAVES_PER_BLOCK 4
#define RAYS_PER_BLOCK  (RAYS_PER_WAVE * WAVES_PER_BLOCK)
#define LDS_STRIDE      17   // 16 rays + 1 pad float: avoids stride-64 bank conflicts

typedef __attribute__((ext_vector_type(2))) float v2f;
typedef __attribute__((ext_vector_type(8))) float v8f;

__global__ __launch_bounds__(WAVES_PER_BLOCK * 32)
void volsdf_weights_kernel(const float* __restrict__ sdf,
                           const float* __restrict__ deltas,
                           const float* __restrict__ beta,
                           float* __restrict__ out)
{
    // Per-wave private staging tile: delta*density in [sample][ray] order.
    __shared__ float tile[WAVES_PER_BLOCK][N_SAMPLES * LDS_STRIDE];

    const int tid   = threadIdx.x;
    const int wave  = tid >> 5;
    const int lane  = tid & 31;
    const int rbase = blockIdx.x * RAYS_PER_BLOCK + wave * RAYS_PER_WAVE;

    const float b     = beta[0];
    const float inv_b = 1.0f / b;

    float* t = tile[wave];

    // ---- Phase 1: coalesced float4 streaming loads; density; transpose to LDS.
    // Wave covers 16*128 = 2048 contiguous floats; iteration `it` == ray `it`.
    const float4* sdf4 = reinterpret_cast<const float4*>(sdf)    + (size_t)rbase * (N_SAMPLES / 4);
    const float4* del4 = reinterpret_cast<const float4*>(deltas) + (size_t)rbase * (N_SAMPLES / 4);
    #pragma unroll
    for (int it = 0; it < RAYS_PER_WAVE; ++it) {
        float4 s4 = sdf4[it * 32 + lane];
        float4 d4 = del4[it * 32 + lane];
        const int samp = lane * 4;
        const float sv[4] = {s4.x, s4.y, s4.z, s4.w};
        const float dv[4] = {d4.x, d4.y, d4.z, d4.w};
        #pragma unroll
        for (int j = 0; j < 4; ++j) {
            // Laplace density, written cancellation-free:
            //   s>=0: 0.5*exp(-s/b)/b        s<0: (1 - 0.5*exp(-|s|/b))/b
            float s = sv[j];
            float e = __expf(-fabsf(s) * inv_b);
            float dens = (s >= 0.0f) ? (0.5f * e * inv_b) : ((1.0f - 0.5f * e) * inv_b);
            t[(samp + j) * LDS_STRIDE + it] = dv[j] * dens;  // dd = delta * density
        }
    }
    __syncthreads();

    // ---- Phase 2: chunked inclusive scan via V_WMMA_F32_16X16X4_F32.
    // I_c(16x16) = L_inc(16x16) * X_c(16x16), done as 4 accumulating K=4 WMMAs.
    // A (16x4 f32): lane&15 -> M, half-wave selects K pair {0,1} or {2,3}.
    // B (4x16 f32): lane&15 -> N (ray), half-wave selects K pair (mirrors A).
    // D (16x16 f32): VGPR v -> M = v (+8 for lanes 16-31), N = lane&15.
    const int n     = lane & 15;            // ray column owned in B/C/D
    const int m     = lane & 15;            // A-matrix row owned by this lane
    const int khalf = (lane >> 4) * 2;      // K sub-pair offset for this half-wave
    const int moff  = (lane >> 4) * 8;      // D rows: VGPR v holds M = v + moff

    // Hoist the triangular A operands: invariant across all 8 chunks.
    v2f aop[4];
    #pragma unroll
    for (int kb = 0; kb < 4; ++kb) {
        const int k0 = 4 * kb + khalf;
        aop[kb].x = (k0     <= m) ? 1.0f : 0.0f;   // L_inc[m][k0]
        aop[kb].y = (k0 + 1 <= m) ? 1.0f : 0.0f;   // L_inc[m][k0+1]
    }

    float running = 0.0f;                   // exclusive carry across chunks (per ray n)
    float* outp = out + (size_t)(rbase + n) * N_SAMPLES;

    #pragma unroll
    for (int c = 0; c < 8; ++c) {
        v8f acc = {};
        #pragma unroll
        for (int kb = 0; kb < 4; ++kb) {
            const int s0 = c * 16 + 4 * kb + khalf;
            v2f bb;
            bb.x = t[s0 * LDS_STRIDE + n];         // X_c[k][n]
            bb.y = t[(s0 + 1) * LDS_STRIDE + n];   // X_c[k+1][n]
            acc = __builtin_amdgcn_wmma_f32_16x16x4_f32(
                false, aop[kb], false, bb, (short)0, acc, false, false);
        }

        // Chunk total per ray = inclusive row M=15 = acc[7] of lane 16+n.
        const float tot = __shfl(acc[7], 16 + n, 32);

        float w[8];
        #pragma unroll
        for (int v = 0; v < 8; ++v) {
            const int mm = v + moff;
            const float x    = t[(c * 16 + mm) * LDS_STRIDE + n]; // dd at (sample,ray)
            const float incl = (c == 0) ? acc[v] : (running + acc[v]);
            const float cs   = incl - x;                          // exclusive cumsum
            const float alpha = 1.0f - __expf(-x);
            const float trans = __expf(-cs);
            w[v] = alpha * trans;
        }
        running = (c == 0) ? tot : (running + tot);

        // Lane's 8 results are 8 consecutive samples of ray n: two aligned float4s.
        float4* o4 = reinterpret_cast<float4*>(outp + c * 16 + moff);
        o4[0] = make_float4(w[0], w[1], w[2], w[3]);
        o4[1] = make_float4(w[4], w[5], w[6], w[7]);
    }
}

extern "C" void kernel_launch(void* const* d_in, const int* in_sizes, int n_in,
                              void* d_out, int out_size, void* d_ws, size_t ws_size,
                              hipStream_t stream) {
    const float* sdf    = (const float*)d_in[0];
    const float* deltas = (const float*)d_in[1];
    const float* beta   = (const float*)d_in[2];
    float* out = (float*)d_out;

    const int n_rays = in_sizes[0] / N_SAMPLES;     // 131072
    const int grid   = n_rays / RAYS_PER_BLOCK;     // 2048 blocks of 128 threads
    volsdf_weights_kernel<<<grid, WAVES_PER_BLOCK * 32, 0, stream>>>(sdf, deltas, beta, out);
}